// dialted_residual_inception_module_down_74972949119509
// MI455X (gfx1250) — compile-verified
//
#include <hip/hip_runtime.h>
#include <hip/hip_bf16.h>

typedef __attribute__((ext_vector_type(2))) float v2f;
typedef __attribute__((ext_vector_type(8))) float v8f;

// Problem dimensions
static constexpr int Bn   = 16;     // batch
static constexpr int Hn   = 128;    // channels
static constexpr int Ln   = 8192;   // sequence length
static constexpr int Nn   = 16;     // SSM states
static constexpr int COn  = 128;    // conv out channels per branch
static constexpr int Tt   = 64;     // chunk length
static constexpr int NCH  = Ln / Tt;   // 128 chunks
static constexpr int LOUT = 2048;      // L/4
static constexpr int NCHAN = 4 * COn;  // 512 output channels

// ---------------------------------------------------------------------------
// K1: precompute per-(h,n) discretization, chunk kernel taps, gather/readout
//     weight tables.
//   Ktap[h][j]   = 2*Re( sum_n CB * w^j ),   j = 0..63
//   G[h][n][u]   = dB * w^(63-u)             (complex, inter-chunk gather)
//   E[h][n][t]   = C  * w^(t+1)              (complex, state readout)
//   wT[h][n]     = w^64                      (complex, state propagation)
// ---------------------------------------------------------------------------
__global__ __launch_bounds__(Tt) void k1_setup(
    const float* __restrict__ log_dt, const float* __restrict__ A_log,
    const float* __restrict__ A_imag, const float* __restrict__ B_re,
    const float* __restrict__ B_im,   const float* __restrict__ C_re,
    const float* __restrict__ C_im,
    float* __restrict__ Ktap, float* __restrict__ G,
    float* __restrict__ E,    float* __restrict__ wT)
{
    int h = blockIdx.x;
    int j = threadIdx.x;                // 0..63
    float dt = expf(log_dt[h]);
    float ks = 0.f;
    for (int n = 0; n < Nn; ++n) {
        int hn = h * Nn + n;
        float Ar = -expf(A_log[hn]);
        float Ai = A_imag[hn];
        float dr = dt * Ar, di = dt * Ai;       // dtA
        // w = exp(dtA)
        float em = expf(dr);
        float wr = em * cosf(di), wi = em * sinf(di);
        // dB = (w - 1)/A * (B_re + i*B_im)
        float er = wr - 1.f, ei = wi;
        float den = Ar * Ar + Ai * Ai;
        float qr = (er * Ar + ei * Ai) / den;
        float qi = (ei * Ar - er * Ai) / den;
        float Br = B_re[hn], Bi = B_im[hn];
        float dBr = qr * Br - qi * Bi;
        float dBi = qr * Bi + qi * Br;
        float Cr = C_re[hn], Ci = C_im[hn];
        float CBr = Cr * dBr - Ci * dBi;
        float CBi = Cr * dBi + Ci * dBr;
        // Ktap term: CB * w^j
        float fj = (float)j;
        float pm = expf(dr * fj);
        float pr = pm * cosf(di * fj), pi = pm * sinf(di * fj);
        ks += 2.f * (CBr * pr - CBi * pi);
        // G[h][n][j] = dB * w^(63-j)
        float fg = (float)(Tt - 1 - j);
        float gm = expf(dr * fg);
        float gr = gm * cosf(di * fg), gi = gm * sinf(di * fg);
        G[(hn * Tt + j) * 2 + 0] = dBr * gr - dBi * gi;
        G[(hn * Tt + j) * 2 + 1] = dBr * gi + dBi * gr;
        // E[h][n][j] = C * w^(j+1)
        float fe = (float)(j + 1);
        float fm = expf(dr * fe);
        float fr = fm * cosf(di * fe), fi = fm * sinf(di * fe);
        E[(hn * Tt + j) * 2 + 0] = Cr * fr - Ci * fi;
        E[(hn * Tt + j) * 2 + 1] = Cr * fi + Ci * fr;
        if (j == 0) {
            float tm = expf(dr * 64.f);
            wT[hn * 2 + 0] = tm * cosf(di * 64.f);
            wT[hn * 2 + 1] = tm * sinf(di * 64.f);
        }
    }
    Ktap[h * Tt + j] = ks;
}

// ---------------------------------------------------------------------------
// K2: inter-chunk state scan. One wave per (b,h); lane n<16 owns complex
// state n. Writes S[b][h][c][n] = state ENTERING chunk c.
//   a_c = sum_u G[n][u] * x[c*64+u] ;  S_{c+1} = w^64 * S_c + a_c
// ---------------------------------------------------------------------------
__global__ __launch_bounds__(256) void k2_scan(
    const float* __restrict__ x, const float* __restrict__ G,
    const float* __restrict__ wT, float* __restrict__ S)
{
    __shared__ float Gs[Nn * Tt * 2];     // 8 KB
    __shared__ float wTs[Nn * 2];
    __shared__ float xs[8][Tt];           // per-wave chunk of x
    int h = blockIdx.x, bg = blockIdx.y;
    int tid = threadIdx.x, wv = tid >> 5, lane = tid & 31;
    int b = bg * 8 + wv;
    for (int i = tid; i < Nn * Tt * 2; i += 256) Gs[i] = G[h * (Nn * Tt * 2) + i];
    if (tid < Nn * 2) wTs[tid] = wT[h * Nn * 2 + tid];
    __syncthreads();

    const float* xb = x + ((size_t)b * Hn + h) * Ln;
    float sr = 0.f, si = 0.f, wtr = 0.f, wti = 0.f;
    if (lane < Nn) { wtr = wTs[lane * 2]; wti = wTs[lane * 2 + 1]; }

    for (int c = 0; c < NCH; ++c) {
        float2 xv = reinterpret_cast<const float2*>(xb + c * Tt)[lane];
        xs[wv][lane * 2]     = xv.x;
        xs[wv][lane * 2 + 1] = xv.y;
        if (c + 1 < NCH)   // hint next chunk into cache: global_prefetch_b8
            __builtin_prefetch(xb + (c + 1) * Tt + lane * 2, 0, 1);
        __builtin_amdgcn_wave_barrier();
        if (lane < Nn) {
            size_t so = ((((size_t)b * Hn + h) * NCH + c) * Nn + lane) * 2;
            S[so] = sr; S[so + 1] = si;
            float ar = 0.f, ai = 0.f;
#pragma unroll 8
            for (int u = 0; u < Tt; ++u) {
                float xu = xs[wv][u];
                ar += Gs[(lane * Tt + u) * 2]     * xu;
                ai += Gs[(lane * Tt + u) * 2 + 1] * xu;
            }
            float nr = wtr * sr - wti * si + ar;
            float ni = wtr * si + wti * sr + ai;
            sr = nr; si = ni;
        }
        __builtin_amdgcn_wave_barrier();
    }
}

// ---------------------------------------------------------------------------
// K3: per-(h,chunk) intra-chunk compute with f32 WMMA.
//   Y(64x16) = TrilToeplitz(Ktap)(64x64) @ X(64x16)
//            + state readout 2*Re(E·S) + D*x
// Block = 128 threads = 4 waves; wave r owns row-block r (one 16x16 D tile).
// ---------------------------------------------------------------------------
__global__ __launch_bounds__(128) void k3_chunk(
    const float* __restrict__ x, const float* __restrict__ Ktap,
    const float* __restrict__ E, const float* __restrict__ S,
    const float* __restrict__ Dv, float* __restrict__ y)
{
    __shared__ float Xs[Tt][16];     // [u][b], conflict-free for WMMA B loads
    __shared__ float KK[80];         // zero-padded taps: KK[16+j] = Ktap[j]
    __shared__ float ErA[Tt][16];    // [t][n]
    __shared__ float EiA[Tt][16];
    __shared__ float SrA[16][16];    // [n][b]
    __shared__ float SiA[16][16];
    __shared__ float Ys[16][68];     // [b][t], stride 68 avoids bank conflicts

    int c = blockIdx.x, h = blockIdx.y;
    int tid = threadIdx.x, lane = tid & 31, wv = tid >> 5;

    {   // stage X chunk (coalesced float4 loads)
        int b = tid >> 3, u0 = (tid & 7) * 8;
        const float4* xp = reinterpret_cast<const float4*>(
            x + ((size_t)b * Hn + h) * Ln + c * Tt + u0);
        float4 p0 = xp[0], p1 = xp[1];
        Xs[u0 + 0][b] = p0.x; Xs[u0 + 1][b] = p0.y;
        Xs[u0 + 2][b] = p0.z; Xs[u0 + 3][b] = p0.w;
        Xs[u0 + 4][b] = p1.x; Xs[u0 + 5][b] = p1.y;
        Xs[u0 + 6][b] = p1.z; Xs[u0 + 7][b] = p1.w;
    }
    if (tid < 16) KK[tid] = 0.f;
    else if (tid < 80) KK[tid] = Ktap[h * Tt + tid - 16];
    for (int i = tid; i < Nn * Tt * 2; i += 128) {
        int n = i >> 7, r = i & 127, t = r >> 1;
        float v = E[h * (Nn * Tt * 2) + i];
        if (r & 1) EiA[t][n] = v; else ErA[t][n] = v;
    }
    for (int i = tid; i < Bn * Nn * 2; i += 128) {
        int b = i >> 5, n = (i >> 1) & 15;
        float v = S[(((size_t)b * Hn + h) * NCH + c) * (Nn * 2) + (i & 31)];
        if (i & 1) SiA[n][b] = v; else SrA[n][b] = v;
    }
    __syncthreads();

    int r = wv;                       // row-block 0..3
    int m = lane & 15;                // N index (batch) / M row within tile
    int hi2 = (lane >> 4) << 1;       // K-sub-offset for upper half-wave
    v8f acc = {0.f, 0.f, 0.f, 0.f, 0.f, 0.f, 0.f, 0.f};
    for (int s = 0; s <= 4 * r + 3; ++s) {   // only K-blocks below diagonal
        int row = r * 16 + m;
        v2f av, bv;
        av[0] = KK[16 + row - (4 * s + hi2)];
        av[1] = KK[16 + row - (4 * s + 1 + hi2)];
        bv[0] = Xs[4 * s + hi2][m];
        bv[1] = Xs[4 * s + 1 + hi2][m];
        acc = __builtin_amdgcn_wmma_f32_16x16x4_f32(
            false, av, false, bv, (short)0, acc, false, false);
    }
    float Dh = Dv[h];
    int hi8 = (lane >> 4) << 3;
#pragma unroll
    for (int j = 0; j < 8; ++j) {
        int t = r * 16 + j + hi8;     // D tile: row = j + 8*(lane>=16)
        float v = acc[j] + Dh * Xs[t][m];
        float ss = 0.f;
#pragma unroll
        for (int n = 0; n < Nn; ++n)
            ss += ErA[t][n] * SrA[n][m] - EiA[t][n] * SiA[n][m];
        Ys[m][t] = v + 2.f * ss;
    }
    __syncthreads();
    {   // coalesced store of y, layout y[h][b][l]
        int b = tid >> 3, t0 = (tid & 7) * 8;
        float4 q0, q1;
        q0.x = Ys[b][t0 + 0]; q0.y = Ys[b][t0 + 1];
        q0.z = Ys[b][t0 + 2]; q0.w = Ys[b][t0 + 3];
        q1.x = Ys[b][t0 + 4]; q1.y = Ys[b][t0 + 5];
        q1.z = Ys[b][t0 + 6]; q1.w = Ys[b][t0 + 7];
        float4* yp = reinterpret_cast<float4*>(
            y + ((size_t)h * Bn + b) * Ln + c * Tt + t0);
        yp[0] = q0; yp[1] = q1;
    }
}

// ---------------------------------------------------------------------------
// K0: zero per-channel stats accumulators (re-run every call: graph replay).
// ---------------------------------------------------------------------------
__global__ void k0_zero(float* __restrict__ stats)
{
    int i = blockIdx.x * 256 + threadIdx.x;
    if (i < NCHAN * 2) stats[i] = 0.f;
}

// ---------------------------------------------------------------------------
// K4: one dilated conv branch as im2col-GEMM with f32 WMMA.
//   out[co, b, lo] = sum_{h,k} W[co,h,k] * y[h, b, lo*4 + (k-1)*dil] + bias
// Fused bias + (branch>0 ? LeakyReLU) and atomic per-channel sum/sumsq.
// Block = 128 threads: one co-tile (16), 4 waves = 4 consecutive lo-tiles.
// ---------------------------------------------------------------------------
__global__ __launch_bounds__(128) void k4_conv(
    const float* __restrict__ y, const float* __restrict__ W,
    const float* __restrict__ bias, int branch, int dil,
    float* __restrict__ out, float* __restrict__ stats)
{
    __shared__ float Ws[Hn * 3][16];   // [kidx][co], 24 KB
    int pg = blockIdx.x, coT = blockIdx.y * 16;
    int tid = threadIdx.x, lane = tid & 31, wv = tid >> 5;
    int b = pg >> 5, loBase = (pg & 31) * 64;

    for (int i = tid; i < Hn * 3 * 16; i += 128) {
        int co = i / (Hn * 3), kk = i - co * (Hn * 3);
        Ws[kk][co] = W[(size_t)(coT + co) * (Hn * 3) + kk];
    }
    __syncthreads();

    int m = lane & 15;                 // column (position) / row-within-tile
    int hi2 = (lane >> 4) << 1;
    int lo = loBase + wv * 16 + m;
    v8f acc = {0.f, 0.f, 0.f, 0.f, 0.f, 0.f, 0.f, 0.f};
    for (int s = 0; s < (Hn * 3) / 4; ++s) {   // 96 K-steps of depth 4
        int k0 = 4 * s + hi2;
        v2f av, bv;
        av[0] = Ws[k0][m];
        av[1] = Ws[k0 + 1][m];
        {
            int hch = k0 / 3, kt = k0 - 3 * hch;
            int li = lo * 4 + (kt - 1) * dil;
            bv[0] = (li >= 0 && li < Ln)
                        ? y[((size_t)hch * Bn + b) * Ln + li] : 0.f;
        }
        {
            int k1 = k0 + 1;
            int hch = k1 / 3, kt = k1 - 3 * hch;
            int li = lo * 4 + (kt - 1) * dil;
            bv[1] = (li >= 0 && li < Ln)
                        ? y[((size_t)hch * Bn + b) * Ln + li] : 0.f;
        }
        acc = __builtin_amdgcn_wmma_f32_16x16x4_f32(
            false, av, false, bv, (short)0, acc, false, false);
    }
    int hi8 = (lane >> 4) << 3;
#pragma unroll
    for (int j = 0; j < 8; ++j) {
        int co = coT + j + hi8;
        float v = acc[j] + bias[co];
        if (branch != 0) v = (v >= 0.f) ? v : 0.3f * v;   // LeakyReLU pre-BN
        out[(size_t)b * ((size_t)NCHAN * LOUT)
            + (size_t)(branch * COn + co) * LOUT + lo] = v;
        // reduce sum / sumsq across the 16 lanes of this row's half-wave
        float s1 = v, s2 = v * v;
        for (int msk = 1; msk < 16; msk <<= 1) {
            s1 += __shfl_xor(s1, msk, 32);
            s2 += __shfl_xor(s2, msk, 32);
        }
        if (m == 0) {
            atomicAdd(&stats[(branch * COn + co) * 2],     s1);
            atomicAdd(&stats[(branch * COn + co) * 2 + 1], s2);
        }
    }
}

// ---------------------------------------------------------------------------
// K5: training-mode BatchNorm applied in place on d_out.
// ---------------------------------------------------------------------------
__global__ __launch_bounds__(256) void k5_bn(
    float* __restrict__ out, const float* __restrict__ stats,
    const float* __restrict__ gamma, const float* __restrict__ beta)
{
    int ch = blockIdx.y;
    int e  = blockIdx.x * 256 + threadIdx.x;     // 0..32767
    int b  = e >> 11, lo = e & 2047;
    const float inv = 1.f / (float)(Bn * LOUT);
    float mean = stats[ch * 2] * inv;
    float var  = stats[ch * 2 + 1] * inv - mean * mean;
    float sc   = gamma[ch & (COn - 1)] * rsqrtf(var + 1e-5f);
    float sh   = beta[ch & (COn - 1)];
    size_t idx = (size_t)b * ((size_t)NCHAN * LOUT) + (size_t)ch * LOUT + lo;
    out[idx] = sc * (out[idx] - mean) + sh;
}

// ---------------------------------------------------------------------------
extern "C" void kernel_launch(void* const* d_in, const int* in_sizes, int n_in,
                              void* d_out, int out_size, void* d_ws, size_t ws_size,
                              hipStream_t stream)
{
    const float* x      = (const float*)d_in[0];
    const float* log_dt = (const float*)d_in[1];
    const float* A_log  = (const float*)d_in[2];
    const float* A_imag = (const float*)d_in[3];
    const float* B_re   = (const float*)d_in[4];
    const float* B_im   = (const float*)d_in[5];
    const float* C_re   = (const float*)d_in[6];
    const float* C_im   = (const float*)d_in[7];
    const float* Dv     = (const float*)d_in[8];
    const float* w[4]  = {(const float*)d_in[9],  (const float*)d_in[11],
                          (const float*)d_in[13], (const float*)d_in[15]};
    const float* bs[4] = {(const float*)d_in[10], (const float*)d_in[12],
                          (const float*)d_in[14], (const float*)d_in[16]};
    const float* gamma = (const float*)d_in[17];
    const float* beta  = (const float*)d_in[18];
    float* out = (float*)d_out;

    // Workspace layout (floats)
    float* ws    = (float*)d_ws;
    float* Ktap  = ws;                                  // H*64
    float* G     = Ktap + (size_t)Hn * Tt;              // H*N*T*2
    float* E     = G + (size_t)Hn * Nn * Tt * 2;        // H*N*T*2
    float* wT    = E + (size_t)Hn * Nn * Tt * 2;        // H*N*2
    float* S     = wT + (size_t)Hn * Nn * 2;            // B*H*NCH*N*2
    float* y     = S + (size_t)Bn * Hn * NCH * Nn * 2;  // H*B*L (layout [h][b][l])
    float* stats = y + (size_t)Hn * Bn * Ln;            // NCHAN*2

    k1_setup<<<Hn, Tt, 0, stream>>>(log_dt, A_log, A_imag, B_re, B_im,
                                    C_re, C_im, Ktap, G, E, wT);
    k2_scan<<<dim3(Hn, Bn / 8), 256, 0, stream>>>(x, G, wT, S);
    k3_chunk<<<dim3(NCH, Hn), 128, 0, stream>>>(x, Ktap, E, S, Dv, y);
    k0_zero<<<(NCHAN * 2 + 255) / 256, 256, 0, stream>>>(stats);
    const int dils[4] = {1, 2, 4, 8};
    for (int i = 0; i < 4; ++i)
        k4_conv<<<dim3((Bn * LOUT) / 64, COn / 16), 128, 0, stream>>>(
            y, w[i], bs[i], i, dils[i], out, stats);
    k5_bn<<<dim3((Bn * LOUT) / 256, NCHAN), 256, 0, stream>>>(out, stats, gamma, beta);
}